// botDetect_48309792145898
// MI455X (gfx1250) — compile-verified
//
#include <hip/hip_runtime.h>
#include <hip/hip_bf16.h>

typedef float v2f __attribute__((ext_vector_type(2)));
typedef float v8f __attribute__((ext_vector_type(8)));

#define FDIM 64

// ---------------------------------------------------------------------------
// K0: init: deg=1 (self loop), agg1=0, d_out seeded with b2 (so the final
// scatter can accumulate straight into it; no separate bias pass).
// ---------------------------------------------------------------------------
__global__ void gcn_init_kernel(float* __restrict__ agg1,
                                float* __restrict__ deg,
                                float* __restrict__ out,
                                const float* __restrict__ b2,
                                int nNodes) {
  int i = blockIdx.x * blockDim.x + threadIdx.x;
  long long total = (long long)nNodes * FDIM;
  if (i < total)      agg1[i] = 0.0f;
  if (i < nNodes)     deg[i]  = 1.0f;             // self loop
  if (i < 2 * nNodes) out[i]  = b2[i & 1];        // bias pre-seeded
}

// ---------------------------------------------------------------------------
// K1: degree accumulation over edge destinations.
// ---------------------------------------------------------------------------
__global__ void gcn_deg_kernel(const int* __restrict__ dstIdx,
                               float* __restrict__ deg, int E) {
  int e = blockIdx.x * blockDim.x + threadIdx.x;
  if (e < E) atomicAdd(&deg[dstIdx[e]], 1.0f);
}

// ---------------------------------------------------------------------------
// K2: dinv = rsqrt(deg), in place (deg >= 1 always).
// ---------------------------------------------------------------------------
__global__ void gcn_rsqrt_kernel(float* __restrict__ deg, int n) {
  int i = blockIdx.x * blockDim.x + threadIdx.x;
  if (i < n) deg[i] = __frsqrt_rn(deg[i]);
}

// ---------------------------------------------------------------------------
// K3: H = X @ W1 with V_WMMA_F32_16X16X4_F32 (full fp32 matrix path).
// Block = 256 threads = 8 waves; each wave owns one 16x16 output tile;
// block covers 32 rows x 64 cols. W1 (64x64, 16KB) staged in LDS.
//
// ISA layouts (7.12.2):
//   A 16x4 f32  : VGPR v, lane l -> M = l%16,      K = v + 2*(l/16)
//   B  4x16 f32 : VGPR v, lane l -> N = l%16,      K = v + 2*(l/16)
//   C/D 16x16   : VGPR r, lane l -> N = l%16,      M = r + 8*(l/16)
// ---------------------------------------------------------------------------
__global__ void gcn_gemm1_wmma_kernel(const float* __restrict__ X,
                                      const float* __restrict__ W,
                                      float* __restrict__ H,
                                      int nRows) {
  __shared__ float Wlds[FDIM * FDIM];
  // stage W1 into LDS: 4096 floats / 256 threads = 16 each (float4 x4)
  for (int i = threadIdx.x * 4; i < FDIM * FDIM; i += blockDim.x * 4) {
    *(float4*)&Wlds[i] = *(const float4*)&W[i];
  }
  __syncthreads();

  const int wave = threadIdx.x >> 5;
  const int lane = threadIdx.x & 31;
  const int m    = lane & 15;
  const int half = lane >> 4;

  const int rowBase = blockIdx.x * 32 + (wave >> 2) * 16;  // 2 row tiles
  const int colBase = (wave & 3) * 16;                     // 4 col tiles
  if (rowBase >= nRows) return;  // wave-uniform: EXEC stays all-ones

  v8f acc = {0.f, 0.f, 0.f, 0.f, 0.f, 0.f, 0.f, 0.f};
  const float* __restrict__ xrow = X + (size_t)(rowBase + m) * FDIM;

#pragma unroll
  for (int k0 = 0; k0 < FDIM; k0 += 4) {
    v2f a, b;
    const int kA = k0 + 2 * half;
    a.x = xrow[kA + 0];                          // K = k0 + 2*half + v
    a.y = xrow[kA + 1];
    b.x = Wlds[(kA + 0) * FDIM + colBase + m];   // same K pattern, N = m
    b.y = Wlds[(kA + 1) * FDIM + colBase + m];
    acc = __builtin_amdgcn_wmma_f32_16x16x4_f32(
        /*neg_a=*/false, a, /*neg_b=*/false, b,
        /*c_mod=*/(short)0, acc, /*reuse_a=*/false, /*reuse_b=*/false);
  }

#pragma unroll
  for (int r = 0; r < 8; ++r) {
    const int mOut = r + 8 * half;
    H[(size_t)(rowBase + mOut) * FDIM + colBase + m] = acc[r];
  }
}

// ---------------------------------------------------------------------------
// K4: layer-1 normalized scatter. One warp (32 lanes) per work item;
// lane handles a float2 slice of the 64-wide feature row. Work items
// [0,E) are edges, [E, E+N) are the self loops.
// agg1[dst] += dinv[src]*dinv[dst] * h[src]
// ---------------------------------------------------------------------------
__global__ void gcn_scatter1_kernel(const int* __restrict__ ei,
                                    const float* __restrict__ dinv,
                                    const float* __restrict__ h,
                                    float* __restrict__ agg1,
                                    int E, int nNodes) {
  long long tid  = (long long)blockIdx.x * blockDim.x + threadIdx.x;
  long long item = tid >> 5;
  int lane = (int)(tid & 31);
  if (item >= (long long)E + nNodes) return;

  int s, d;
  if (item < E) { s = ei[item]; d = ei[(long long)E + item]; }
  else          { s = d = (int)(item - E); }

  float norm = dinv[s] * dinv[d];
  float2 v = *(const float2*)(h + (size_t)s * FDIM + lane * 2);
  float* o = agg1 + (size_t)d * FDIM + lane * 2;
  atomicAdd(o + 0, v.x * norm);
  atomicAdd(o + 1, v.y * norm);
}

// ---------------------------------------------------------------------------
// K5: fused bias + ReLU + layer-2 GEMM (64 -> 2). One thread per node.
// h2[n] = relu(agg1[n] + b1) @ W2
// ---------------------------------------------------------------------------
__global__ void gcn_layer2_kernel(const float* __restrict__ agg1,
                                  const float* __restrict__ b1,
                                  const float* __restrict__ W2,
                                  float* __restrict__ h2, int nNodes) {
  int n = blockIdx.x * blockDim.x + threadIdx.x;
  if (n >= nNodes) return;
  const float* a = agg1 + (size_t)n * FDIM;
  float s0 = 0.f, s1 = 0.f;
#pragma unroll
  for (int f = 0; f < FDIM; ++f) {
    float v = a[f] + b1[f];
    v = v > 0.f ? v : 0.f;
    s0 = fmaf(v, W2[2 * f + 0], s0);
    s1 = fmaf(v, W2[2 * f + 1], s1);
  }
  h2[2 * n + 0] = s0;
  h2[2 * n + 1] = s1;
}

// ---------------------------------------------------------------------------
// K6: layer-2 scatter into d_out (pre-seeded with b2). One thread per
// work item (edge or self loop), 2 output channels.
// ---------------------------------------------------------------------------
__global__ void gcn_scatter2_kernel(const int* __restrict__ ei,
                                    const float* __restrict__ dinv,
                                    const float* __restrict__ h2,
                                    float* __restrict__ out,
                                    int E, int nNodes) {
  long long item = (long long)blockIdx.x * blockDim.x + threadIdx.x;
  if (item >= (long long)E + nNodes) return;

  int s, d;
  if (item < E) { s = ei[item]; d = ei[(long long)E + item]; }
  else          { s = d = (int)(item - E); }

  float norm = dinv[s] * dinv[d];
  float2 v = *(const float2*)(h2 + (size_t)s * 2);
  atomicAdd(&out[(size_t)d * 2 + 0], v.x * norm);
  atomicAdd(&out[(size_t)d * 2 + 1], v.y * norm);
}

// ---------------------------------------------------------------------------
extern "C" void kernel_launch(void* const* d_in, const int* in_sizes, int n_in,
                              void* d_out, int out_size, void* d_ws, size_t ws_size,
                              hipStream_t stream) {
  const float* x  = (const float*)d_in[0];
  const int*   ei = (const int*)  d_in[1];   // edge_index [2, E], int32
  const float* W1 = (const float*)d_in[2];
  const float* b1 = (const float*)d_in[3];
  const float* W2 = (const float*)d_in[4];
  const float* b2 = (const float*)d_in[5];
  float* out = (float*)d_out;

  const int nNodes = in_sizes[0] / FDIM;     // 100000
  const int E      = in_sizes[1] / 2;        // 1600000

  // workspace carve-out (floats): deg | h | agg1 | h2
  float* ws   = (float*)d_ws;
  float* deg  = ws;
  float* h    = deg  + nNodes;
  float* agg1 = h    + (size_t)nNodes * FDIM;
  float* h2   = agg1 + (size_t)nNodes * FDIM;

  const int TB = 256;

  // K0: init (covers nNodes*64 elements -> also covers nNodes and 2*nNodes)
  {
    long long total = (long long)nNodes * FDIM;
    int grid = (int)((total + TB - 1) / TB);
    gcn_init_kernel<<<grid, TB, 0, stream>>>(agg1, deg, out, b2, nNodes);
  }
  // K1: degree
  gcn_deg_kernel<<<(E + TB - 1) / TB, TB, 0, stream>>>(ei + E, deg, E);
  // K2: dinv
  gcn_rsqrt_kernel<<<(nNodes + TB - 1) / TB, TB, 0, stream>>>(deg, nNodes);
  // K3: WMMA GEMM  h = X @ W1   (32 rows per block)
  gcn_gemm1_wmma_kernel<<<(nNodes + 31) / 32, TB, 0, stream>>>(x, W1, h, nNodes);
  // K4: layer-1 scatter (warp per edge/self-loop)
  {
    long long work = ((long long)E + nNodes) * 32;
    int grid = (int)((work + TB - 1) / TB);
    gcn_scatter1_kernel<<<grid, TB, 0, stream>>>(ei, deg, h, agg1, E, nNodes);
  }
  // K5: fused bias+ReLU+W2
  gcn_layer2_kernel<<<(nNodes + TB - 1) / TB, TB, 0, stream>>>(agg1, b1, W2, h2, nNodes);
  // K6: layer-2 scatter into pre-biased d_out
  {
    long long work = (long long)E + nNodes;
    int grid = (int)((work + TB - 1) / TB);
    gcn_scatter2_kernel<<<grid, TB, 0, stream>>>(ei, deg, h2, out, E, nNodes);
  }
}